// Quantumnet_63754494542393
// MI455X (gfx1250) — compile-verified
//
#include <hip/hip_runtime.h>
#include <math.h>

typedef __attribute__((ext_vector_type(2))) float v2f;
typedef __attribute__((ext_vector_type(8))) float v8f;

#define NQ      12
#define DIM     4096
#define NLAYERS 20
#define F_IN    2048
#define PROW    17          // 16-float rows padded to 17 (bank-conflict free)
#define GROW    (16 * PROW) // padded size of one a0-slab (272 elements)

// padded LDS index for flat state index i (i = a0*256 + a1*16 + a2)
__device__ __forceinline__ int pidx(int i) { return (i >> 4) * PROW + (i & 15); }

// Source index of the fixed entangling-layer permutation: new[i] = old[cnot_src(i)].
// CNOT application order: (0,1),(2,3),(4,5),(6,7),(8,9),(10,11), (1,2),(3,4),(5,6),(7,8),(9,10).
// state_n[i] = state_0[g_1(g_2(...g_n(i)))] -> apply involutions in REVERSE order.
// Qubit q lives at bit (11-q) of the flat index (qubit 0 = most significant axis).
__device__ __forceinline__ int cnot_src(int i) {
    const int cq[11] = {0, 2, 4, 6, 8, 10, 1, 3, 5, 7, 9};
    const int tq[11] = {1, 3, 5, 7, 9, 11, 2, 4, 6, 8, 10};
    int j = i;
#pragma unroll
    for (int k = 10; k >= 0; --k) {
        int cm = 1 << (11 - cq[k]);
        int tm = 1 << (11 - tq[k]);
        if (j & cm) j ^= tm;
    }
    return j;
}

// tanh(x) * pi/2 via hardware v_exp/v_rcp; clamp keeps exp finite (tanh(+-15)==+-1 in f32)
__device__ __forceinline__ float tanh_pio2(float x) {
    float xc = fminf(fmaxf(x, -15.f), 15.f);
    float z  = __expf(xc * 2.0f);
    return (z - 1.0f) * __builtin_amdgcn_rcpf(z + 1.0f) * 1.57079632679489662f;
}

__global__ __launch_bounds__(256)
void qnet_kernel(const float* __restrict__ in, const float* __restrict__ qp,
                 float* __restrict__ out)
{
    __shared__ float Sa[256 * PROW];   // state buffer A (padded)
    __shared__ float Sb[256 * PROW];   // state buffer B (padded)
    __shared__ float Mm[3][16 * PROW]; // 16x16 group rotation matrices (padded rows)
    __shared__ float CS[NQ], SN[NQ];   // cos/sin of theta/2 for current layer
    __shared__ float red[96];          // cross-wave reduction scratch
    __shared__ float s_scale;

    const int t    = threadIdx.x; // 0..255
    const int lane = t & 31;
    const int wave = t >> 5;      // 0..7
    const int hi   = lane >> 4;   // 0/1 : lane half (WMMA layout)
    const int lo   = lane & 15;   // M (A) / N (B,C,D) index
    const int row  = blockIdx.x;

    // ---------------- amplitude embedding ----------------
    const float* srcrow = in + (size_t)row * F_IN;
    float ss = 0.f;
#pragma unroll
    for (int e = 0; e < 8; ++e) {
        int i = e * 256 + t;
        float q = tanh_pio2(srcrow[i]);
        Sa[pidx(i)] = q;
        ss += q * q;
    }
#pragma unroll
    for (int e = 8; e < 16; ++e) Sa[pidx(e * 256 + t)] = 0.5f; // padding value

    for (int off = 16; off; off >>= 1) ss += __shfl_down(ss, off, 32);
    if (lane == 0) red[wave] = ss;
    __syncthreads();
    if (t == 0) {
        float tot = 512.0f; // 2048 pad elements * 0.25
        for (int w = 0; w < 8; ++w) tot += red[w];
        s_scale = 1.0f / sqrtf(tot);
    }
    __syncthreads();
    {
        float sc = s_scale;
#pragma unroll
        for (int e = 0; e < 16; ++e) Sa[pidx(e * 256 + t)] *= sc;
    }

    // Precompute permuted A-fragment LDS addresses for the axis-2 matmul.
    // Axis-2 A tile row r = T*16+lo, element k = 4s+2*hi+d.
    // The CNOT permutation is folded into these reads -> entangling layer is free.
    int srcA[16];
#pragma unroll
    for (int tt = 0; tt < 2; ++tt) {
        int T = wave + tt * 8;
#pragma unroll
        for (int s = 0; s < 4; ++s) {
#pragma unroll
            for (int d = 0; d < 2; ++d) {
                int i = (T * 16 + lo) * 16 + 4 * s + 2 * hi + d;
                srcA[tt * 8 + s * 2 + d] = pidx(cnot_src(i));
            }
        }
    }

    float* pa = Sa; // current state
    float* pb = Sb;

    const int T0 = wave;      // first tile handled by this wave
    const int T1 = wave + 8;  // second tile

    const float* th = qp + NQ; // scan uses q_params rows 1..20

    for (int L = 0; L < NLAYERS; ++L) {
        if (t < NQ) {
            float a = th[L * NQ + t] * 0.5f;
            CS[t] = cosf(a);
            SN[t] = sinf(a);
        }
        __syncthreads();

        // Build M_g = kron(RY(t_{4g}),...,RY(t_{4g+3})), one entry per thread per group.
        {
            int r = t >> 4, c = t & 15;
#pragma unroll
            for (int g = 0; g < 3; ++g) {
                float f = 1.f;
#pragma unroll
                for (int p = 0; p < 4; ++p) {
                    int q  = g * 4 + p;
                    int rb = (r >> (3 - p)) & 1;
                    int cb = (c >> (3 - p)) & 1;
                    f *= (rb == cb) ? CS[q] : (rb ? SN[q] : -SN[q]);
                }
                Mm[g][r * PROW + c] = f;
            }
        }
        __syncthreads();

        // ---- entangle (folded into reads) + RY on qubits 8..11 : axis-2 ----
        // D[r][c] = sum_k Sperm[r*16+k] * M2[c][k]  (A = permuted state, B = M2^T, shared)
        {
            v8f acc0 = {0, 0, 0, 0, 0, 0, 0, 0};
            v8f acc1 = {0, 0, 0, 0, 0, 0, 0, 0};
#pragma unroll
            for (int s = 0; s < 4; ++s) {
                int k0 = 4 * s + 2 * hi;
                v2f a0, a1, b;
                a0.x = pa[srcA[s * 2 + 0]];
                a0.y = pa[srcA[s * 2 + 1]];
                a1.x = pa[srcA[8 + s * 2 + 0]];
                a1.y = pa[srcA[8 + s * 2 + 1]];
                b.x  = Mm[2][lo * PROW + k0];     // B[k][n] = M2[n][k]
                b.y  = Mm[2][lo * PROW + k0 + 1];
                acc0 = __builtin_amdgcn_wmma_f32_16x16x4_f32(
                    false, a0, false, b, (short)0, acc0, false, false);
                acc1 = __builtin_amdgcn_wmma_f32_16x16x4_f32(
                    false, a1, false, b, (short)0, acc1, false, false);
            }
#pragma unroll
            for (int v = 0; v < 8; ++v) {
                pb[(T0 * 16 + v + 8 * hi) * PROW + lo] = acc0[v];
                pb[(T1 * 16 + v + 8 * hi) * PROW + lo] = acc1[v];
            }
        }
        __syncthreads();

        // ---- RY on qubits 4..7 : axis-1 (per a0 slab: D = M1 * S_a0; A shared) ----
        {
            v8f acc0 = {0, 0, 0, 0, 0, 0, 0, 0};
            v8f acc1 = {0, 0, 0, 0, 0, 0, 0, 0};
#pragma unroll
            for (int s = 0; s < 4; ++s) {
                int k0 = 4 * s + 2 * hi;
                v2f a, b0, b1;
                a.x  = Mm[1][lo * PROW + k0];
                a.y  = Mm[1][lo * PROW + k0 + 1];
                b0.x = pb[T0 * GROW + k0 * PROW + lo];
                b0.y = pb[T0 * GROW + (k0 + 1) * PROW + lo];
                b1.x = pb[T1 * GROW + k0 * PROW + lo];
                b1.y = pb[T1 * GROW + (k0 + 1) * PROW + lo];
                acc0 = __builtin_amdgcn_wmma_f32_16x16x4_f32(
                    false, a, false, b0, (short)0, acc0, false, false);
                acc1 = __builtin_amdgcn_wmma_f32_16x16x4_f32(
                    false, a, false, b1, (short)0, acc1, false, false);
            }
#pragma unroll
            for (int v = 0; v < 8; ++v) {
                pa[T0 * GROW + (v + 8 * hi) * PROW + lo] = acc0[v];
                pa[T1 * GROW + (v + 8 * hi) * PROW + lo] = acc1[v];
            }
        }
        __syncthreads();

        // ---- RY on qubits 0..3 : axis-0 (D = M0 * S viewed as 16x256; A shared) ----
        {
            v8f acc0 = {0, 0, 0, 0, 0, 0, 0, 0};
            v8f acc1 = {0, 0, 0, 0, 0, 0, 0, 0};
#pragma unroll
            for (int s = 0; s < 4; ++s) {
                int k0 = 4 * s + 2 * hi;
                v2f a, b0, b1;
                a.x  = Mm[0][lo * PROW + k0];
                a.y  = Mm[0][lo * PROW + k0 + 1];
                b0.x = pa[k0 * GROW + T0 * PROW + lo];
                b0.y = pa[(k0 + 1) * GROW + T0 * PROW + lo];
                b1.x = pa[k0 * GROW + T1 * PROW + lo];
                b1.y = pa[(k0 + 1) * GROW + T1 * PROW + lo];
                acc0 = __builtin_amdgcn_wmma_f32_16x16x4_f32(
                    false, a, false, b0, (short)0, acc0, false, false);
                acc1 = __builtin_amdgcn_wmma_f32_16x16x4_f32(
                    false, a, false, b1, (short)0, acc1, false, false);
            }
#pragma unroll
            for (int v = 0; v < 8; ++v) {
                pb[(v + 8 * hi) * GROW + T0 * PROW + lo] = acc0[v];
                pb[(v + 8 * hi) * GROW + T1 * PROW + lo] = acc1[v];
            }
        }
        __syncthreads();

        // layer net effect: pa -> pb ; swap for next layer
        float* tmp = pa; pa = pb; pb = tmp;
    }

    // ---------------- readout: <Z_q> = sum_i sign_q(i) * state[i]^2 ----------------
    // i = e*256 + t: qubits 0..3 (bits 11..8) depend only on compile-time e;
    // qubits 4..11 (bits 7..0) depend only on t -> one total sum P + per-thread sign.
    float P = 0.f;                 // plain sum of probs (for qubits 4..11)
    float A[4] = {0.f, 0.f, 0.f, 0.f}; // signed sums for qubits 0..3
#pragma unroll
    for (int e = 0; e < 16; ++e) {
        int i = e * 256 + t;
        float v = pa[pidx(i)];
        float p = v * v;
        P += p;
#pragma unroll
        for (int q = 0; q < 4; ++q)
            A[q] += ((e >> (3 - q)) & 1) ? -p : p; // compile-time sign
    }
    float acc[NQ];
#pragma unroll
    for (int q = 0; q < 4; ++q) acc[q] = A[q];
#pragma unroll
    for (int q = 4; q < NQ; ++q)
        acc[q] = ((t >> (11 - q)) & 1) ? -P : P;   // runtime per-thread sign

    for (int off = 16; off; off >>= 1) {
#pragma unroll
        for (int q = 0; q < NQ; ++q) acc[q] += __shfl_down(acc[q], off, 32);
    }
    if (lane == 0) {
#pragma unroll
        for (int q = 0; q < NQ; ++q) red[wave * NQ + q] = acc[q];
    }
    __syncthreads();
    if (t < NQ) {
        float s = 0.f;
#pragma unroll
        for (int w = 0; w < 8; ++w) s += red[w * NQ + t];
        out[(size_t)row * NQ + t] = s;
    }
}

extern "C" void kernel_launch(void* const* d_in, const int* in_sizes, int n_in,
                              void* d_out, int out_size, void* d_ws, size_t ws_size,
                              hipStream_t stream) {
    (void)n_in; (void)out_size; (void)d_ws; (void)ws_size;
    const float* in  = (const float*)d_in[0]; // (4096, 2048) f32
    const float* qp  = (const float*)d_in[1]; // (252,) f32
    float*       out = (float*)d_out;         // (4096, 12) f32
    int rows = in_sizes[0] / F_IN;            // 4096
    qnet_kernel<<<rows, 256, 0, stream>>>(in, qp, out);
}